// GNN_31190052503646
// MI455X (gfx1250) — compile-verified
//
#include <hip/hip_runtime.h>
#include <hip/hip_bf16.h>
#include <cstdint>

typedef float v2f __attribute__((ext_vector_type(2)));
typedef float v8f __attribute__((ext_vector_type(8)));

#define N_NODES 10000

// ---------------------------------------------------------------------------
// Degree / normalization kernels
// ---------------------------------------------------------------------------
__global__ void deg_init_kernel(float* __restrict__ deg, int n) {
    int i = blockIdx.x * blockDim.x + threadIdx.x;
    if (i < n) deg[i] = 1.0f;  // +1 self-loop
}

__global__ void deg_count_kernel(const int* __restrict__ dst, float* __restrict__ deg, int E) {
    int e = blockIdx.x * blockDim.x + threadIdx.x;
    if (e < E) unsafeAtomicAdd(&deg[dst[e]], 1.0f);
}

__global__ void dinv_kernel(const float* __restrict__ deg, float* __restrict__ dinv, int n) {
    int i = blockIdx.x * blockDim.x + threadIdx.x;
    if (i < n) dinv[i] = rsqrtf(deg[i]);
}

// ---------------------------------------------------------------------------
// FP32 WMMA GEMM: C[M,N] = A[M,K] * B[K,N], row-major.
// One wave computes a 16x64 output tile via 4 accumulators, so each A 16x4
// fragment load feeds 4 back-to-back V_WMMA_F32_16X16X4_F32 issues.
// M%16==0, K%4==0, N%64==0 assumed (holds: M=10000, K in {256,512}, N in {512,256}).
//
// A 16x4 layout (ISA 7.12.2, 32-bit A):  lanes 0-15 : row=lane,    v.x=K0, v.y=K1
//                                        lanes 16-31: row=lane-16, v.x=K2, v.y=K3
// B 4x16 layout (rows striped across lanes): v.x = row (k+2*half), v.y = row+1
// C/D 16x16  : VGPR v -> row M = v + 8*half, col N = lane&15
// ---------------------------------------------------------------------------
__global__ void __launch_bounds__(32)
gemm_wmma_f32_kernel(const float* __restrict__ A, const float* __restrict__ B,
                     float* __restrict__ C, int M, int K, int N) {
    const int m0   = blockIdx.x << 4;
    const int n0   = blockIdx.y << 6;     // 64 columns per wave
    const int lane = threadIdx.x;         // 0..31
    const int half = lane >> 4;           // 0 | 1
    const int l16  = lane & 15;

    v8f acc0 = {}, acc1 = {}, acc2 = {}, acc3 = {};
    const float* arow  = A + (size_t)(m0 + l16) * K + (half << 1);
    const float* bbase = B + (size_t)(half << 1) * N + n0 + l16;

    for (int k = 0; k < K; k += 4) {
        v2f a;
        a.x = arow[k + 0];                 // contiguous pair -> global_load_b64
        a.y = arow[k + 1];
        const float* brow0 = bbase + (size_t)k * N;  // row k + 2*half
        const float* brow1 = brow0 + N;              // row k + 2*half + 1
        v2f b0, b1, b2, b3;
        b0.x = brow0[0];  b0.y = brow1[0];
        b1.x = brow0[16]; b1.y = brow1[16];
        b2.x = brow0[32]; b2.y = brow1[32];
        b3.x = brow0[48]; b3.y = brow1[48];
        acc0 = __builtin_amdgcn_wmma_f32_16x16x4_f32(false, a, false, b0, (short)0, acc0, false, false);
        acc1 = __builtin_amdgcn_wmma_f32_16x16x4_f32(false, a, false, b1, (short)0, acc1, false, false);
        acc2 = __builtin_amdgcn_wmma_f32_16x16x4_f32(false, a, false, b2, (short)0, acc2, false, false);
        acc3 = __builtin_amdgcn_wmma_f32_16x16x4_f32(false, a, false, b3, (short)0, acc3, false, false);
    }

    float* crow = C + (size_t)(m0 + (half << 3)) * N + n0 + l16;
#pragma unroll
    for (int v = 0; v < 8; ++v) {
        float* cr = crow + (size_t)v * N;
        cr[0]  = acc0[v];
        cr[16] = acc1[v];
        cr[32] = acc2[v];
        cr[48] = acc3[v];
    }
}

// ---------------------------------------------------------------------------
// agg[i][f] = xw[i][f] * dinv[i]^2 + b[f]   (self-loop term + bias, float4)
// ---------------------------------------------------------------------------
__global__ void init_agg_kernel(const float* __restrict__ xw, const float* __restrict__ dinv,
                                const float* __restrict__ b, float* __restrict__ agg,
                                int n_nodes, int H) {
    int idx4 = blockIdx.x * blockDim.x + threadIdx.x;      // index in float4 units
    int total4 = (n_nodes * H) >> 2;
    if (idx4 >= total4) return;
    int idx  = idx4 << 2;
    int node = idx / H;
    int f    = idx - node * H;
    float s  = dinv[node];
    s = s * s;
    float4 v  = *(const float4*)(xw + idx);
    float4 bv = *(const float4*)(b + f);
    float4 r;
    r.x = v.x * s + bv.x;
    r.y = v.y * s + bv.y;
    r.z = v.z * s + bv.z;
    r.w = v.w * s + bv.w;
    *(float4*)(agg + idx) = r;
}

// ---------------------------------------------------------------------------
// Edge scatter: agg[dst[e]][:] += xw[src[e]][:] * dinv[src]*dinv[dst]
// One block per edge; threads cover features in float4 chunks. Gathers hit L2
// (xw fits in 192MB L2); atomics are non-returning global_atomic_add_f32.
// ---------------------------------------------------------------------------
__global__ void scatter_edges_kernel(const float* __restrict__ xw,
                                     const int* __restrict__ src,
                                     const int* __restrict__ dst,
                                     const float* __restrict__ dinv,
                                     float* __restrict__ agg, int H) {
    const int e = blockIdx.x;
    const int s = src[e];
    const int d = dst[e];
    const float norm = dinv[s] * dinv[d];
    const float4* xr = (const float4*)(xw + (size_t)s * H);
    float* ar = agg + (size_t)d * H;
    const int H4 = H >> 2;
    for (int f4 = threadIdx.x; f4 < H4; f4 += blockDim.x) {
        float4 v = xr[f4];
        float* a = ar + (f4 << 2);
        unsafeAtomicAdd(a + 0, v.x * norm);
        unsafeAtomicAdd(a + 1, v.y * norm);
        unsafeAtomicAdd(a + 2, v.z * norm);
        unsafeAtomicAdd(a + 3, v.w * norm);
    }
}

// ---------------------------------------------------------------------------
// In-place ReLU (float4)
// ---------------------------------------------------------------------------
__global__ void relu_kernel(float* __restrict__ x, int total4) {
    int i = blockIdx.x * blockDim.x + threadIdx.x;
    if (i >= total4) return;
    float4 v = *(float4*)(x + (i << 2));
    v.x = fmaxf(v.x, 0.0f);
    v.y = fmaxf(v.y, 0.0f);
    v.z = fmaxf(v.z, 0.0f);
    v.w = fmaxf(v.w, 0.0f);
    *(float4*)(x + (i << 2)) = v;
}

// ---------------------------------------------------------------------------
// Fused ReLU + global max-pool: out[f] = max_i relu(agg[i][f])
// One block per feature column; LDS tree reduction. relu >= 0 so init = 0.
// ---------------------------------------------------------------------------
__global__ void maxpool_kernel(const float* __restrict__ agg, float* __restrict__ out,
                               int n_nodes, int F) {
    const int f = blockIdx.x;
    float m = 0.0f;
    for (int i = threadIdx.x; i < n_nodes; i += blockDim.x)
        m = fmaxf(m, agg[(size_t)i * F + f]);  // relu folded: max(·,0) via init 0
    __shared__ float red[256];
    red[threadIdx.x] = m;
    __syncthreads();
    for (int s = blockDim.x >> 1; s > 0; s >>= 1) {
        if (threadIdx.x < (unsigned)s)
            red[threadIdx.x] = fmaxf(red[threadIdx.x], red[threadIdx.x + s]);
        __syncthreads();
    }
    if (threadIdx.x == 0) out[f] = red[0];
}

// ---------------------------------------------------------------------------
// Launcher
// ---------------------------------------------------------------------------
extern "C" void kernel_launch(void* const* d_in, const int* in_sizes, int n_in,
                              void* d_out, int out_size, void* d_ws, size_t ws_size,
                              hipStream_t stream) {
    const float* x  = (const float*)d_in[0];
    const int*   ei = (const int*)d_in[1];
    const float* W1 = (const float*)d_in[2];
    const float* b1 = (const float*)d_in[3];
    const float* W2 = (const float*)d_in[4];
    const float* b2 = (const float*)d_in[5];
    float* out = (float*)d_out;

    const int N = N_NODES;
    const int F = in_sizes[5];              // 256 (len of b2)
    const int H = in_sizes[3];              // 512 (len of b1)
    const int E = in_sizes[1] / 2;          // 320000
    const int* src = ei;
    const int* dst = ei + E;

    // workspace carve-up (floats)
    float* ws   = (float*)d_ws;
    float* deg  = ws;                        // N
    float* dinv = deg  + N;                  // N
    float* xw1  = dinv + N;                  // N*H
    float* agg1 = xw1  + (size_t)N * H;      // N*H
    float* xw2  = agg1 + (size_t)N * H;      // N*F
    float* agg2 = xw2  + (size_t)N * F;      // N*F

    // 1) degrees on A + I, dinv = rsqrt(deg)
    deg_init_kernel<<<(N + 255) / 256, 256, 0, stream>>>(deg, N);
    deg_count_kernel<<<(E + 255) / 256, 256, 0, stream>>>(dst, deg, E);
    dinv_kernel<<<(N + 255) / 256, 256, 0, stream>>>(deg, dinv, N);

    // 2) Layer 1: xw1 = x @ W1  (M=N, K=F, N=H) via fp32 WMMA, 16x64 tiles
    gemm_wmma_f32_kernel<<<dim3(N / 16, H / 64), 32, 0, stream>>>(x, W1, xw1, N, F, H);
    // agg1 = xw1*dinv^2 + b1 ; += scattered messages ; relu in place
    init_agg_kernel<<<((N * H / 4) + 255) / 256, 256, 0, stream>>>(xw1, dinv, b1, agg1, N, H);
    scatter_edges_kernel<<<E, 128, 0, stream>>>(xw1, src, dst, dinv, agg1, H);
    relu_kernel<<<((N * H / 4) + 255) / 256, 256, 0, stream>>>(agg1, N * H / 4);

    // 3) Layer 2: xw2 = h1 @ W2  (M=N, K=H, N=F)
    gemm_wmma_f32_kernel<<<dim3(N / 16, F / 64), 32, 0, stream>>>(agg1, W2, xw2, N, H, F);
    init_agg_kernel<<<((N * F / 4) + 255) / 256, 256, 0, stream>>>(xw2, dinv, b2, agg2, N, F);
    scatter_edges_kernel<<<E, 64, 0, stream>>>(xw2, src, dst, dinv, agg2, F);

    // 4) fused relu + global max pool -> (1, F)
    maxpool_kernel<<<F, 256, 0, stream>>>(agg2, out, N, F);
}